// MXFPOptimizer_35459249996636
// MI455X (gfx1250) — compile-verified
//
#include <hip/hip_runtime.h>
#include <stdint.h>

// MXFP6 (BLOCK=32, MAX_MANTISSA=15) quant-dequant; STE forward == dequant value.
// Pure streaming: 128 MiB in + 128 MiB out -> ~11.5 us floor at 23.3 TB/s.
//
// Mapping: each lane owns 16 CONSECUTIVE elements (4x float4, 64B) == half an
// MX block; lanes (2k, 2k+1) of a wave32 share one 32-element block. Block
// work (abs-max reduce, exponent bit-trick, one IEEE divide) is amortized over
// 16 elements, and the cross-lane reduce is a single ds_swizzle XOR-1 swap
// (LDS crossbar only, no LDS storage, no barrier).
//
// NT cache hints on both directions: the 256 MB stream exceeds the 192 MB L2
// and each byte is touched exactly once.
//
// Exact launch: 33,554,432 elems / 16 per lane = 2,097,152 threads = 8192x256.
// No loop, no divergence (n is divisible by 16 and by the block size 32).

typedef float v4f __attribute__((ext_vector_type(4)));

#define SWZ_XOR1 0x041F   // group-of-32 mode: xor_mask=1, and_mask=0x1f

__device__ __forceinline__ float vmax4_abs(v4f v) {
    v4f a = __builtin_elementwise_abs(v);
    return fmaxf(fmaxf(a.x, a.y), fmaxf(a.z, a.w));   // folds to v_max3 + |mods|
}

__device__ __forceinline__ v4f qdq4(v4f v, float r15, float ds) {
    // q = clip(rne(v * 15/scale), -15, 15); out = q * (scale/15)
    v4f t = v * r15;
    v4f q = __builtin_elementwise_roundeven(t);       // v_rndne (RNE, matches jnp.round)
    q = __builtin_elementwise_max(q, (v4f)(-15.0f));
    q = __builtin_elementwise_min(q, (v4f)( 15.0f));
    return q * ds;
}

__global__ __launch_bounds__(256)
void mxfp6_qdq_kernel(const float* __restrict__ x, float* __restrict__ y, int n16) {
    int t = blockIdx.x * blockDim.x + threadIdx.x;    // chunk id, 16 elems per chunk
    if (t >= n16) return;

    const v4f* __restrict__ px = (const v4f*)x + (size_t)t * 4;
    v4f a = __builtin_nontemporal_load(px + 0);       // global_load_b128 th:TH_LOAD_NT
    v4f b = __builtin_nontemporal_load(px + 1);
    v4f c = __builtin_nontemporal_load(px + 2);
    v4f d = __builtin_nontemporal_load(px + 3);

    // In-lane abs-max over 16 elements (max3 tree with abs source modifiers).
    float m = fmaxf(fmaxf(vmax4_abs(a), vmax4_abs(b)),
                    fmaxf(vmax4_abs(c), vmax4_abs(d)));

    // Partner lane holds the other half of this 32-element MX block.
    m = fmaxf(m, __int_as_float(__builtin_amdgcn_ds_swizzle(__float_as_int(m), SWZ_XOR1)));
    m = fmaxf(m, 1e-8f);

    // scale = 2^floor(log2(m)) == m with mantissa zeroed (m is normal, > 0).
    uint32_t sb   = __float_as_uint(m) & 0xFF800000u;
    float scale   = __uint_as_float(sb);
    // 1/scale via exponent negation (exact), *15 (exact: 15 * 2^-e).
    float r15     = __uint_as_float(0x7F000000u - sb) * 15.0f;  // 15/scale
    float ds      = scale / 15.0f;   // one IEEE divide per 16 elements

    v4f* __restrict__ py = (v4f*)y + (size_t)t * 4;
    __builtin_nontemporal_store(qdq4(a, r15, ds), py + 0);      // global_store_b128 NT
    __builtin_nontemporal_store(qdq4(b, r15, ds), py + 1);
    __builtin_nontemporal_store(qdq4(c, r15, ds), py + 2);
    __builtin_nontemporal_store(qdq4(d, r15, ds), py + 3);
}

extern "C" void kernel_launch(void* const* d_in, const int* in_sizes, int n_in,
                              void* d_out, int out_size, void* d_ws, size_t ws_size,
                              hipStream_t stream) {
    const float* x = (const float*)d_in[0];
    float* y = (float*)d_out;
    const long long n = (long long)in_sizes[0];   // 4096*8192 = 33,554,432
    const int n16 = (int)(n / 16);                // 2,097,152 chunks of 16 elems

    const int block = 256;                        // 8 wave32s
    const int grid  = (n16 + block - 1) / block;  // 8192 blocks, exact fit
    mxfp6_qdq_kernel<<<grid, block, 0, stream>>>(x, y, n16);
}